// Embedding_70342974374205
// MI455X (gfx1250) — compile-verified
//
#include <hip/hip_runtime.h>

// Embedding lookup (one_hot @ weight == row gather) for MI455X / gfx1250.
// Strategy: TDM gather-mode DMA (tensor_load_to_lds) pulls 16 arbitrary
// weight rows (16 x 768 f32 = 48 KB) into LDS per workgroup; then all waves
// stream the contiguous tile to the contiguous output slab with b128 ops.
// Pure bandwidth problem (~50 MB traffic, ~2.2 us at 23.3 TB/s) -> no WMMA.

typedef unsigned int       u32;
typedef unsigned long long u64;
typedef __attribute__((ext_vector_type(4))) u32   v4u;
typedef __attribute__((ext_vector_type(4))) int   v4i;
typedef __attribute__((ext_vector_type(8))) int   v8i;
typedef __attribute__((ext_vector_type(4))) float vf4;

#define ROW            768            // floats per embedding row
#define TOK_PER_BLK    16             // rows gathered per TDM descriptor (16-bit idx mode max)
#define BLK_THREADS    256            // 8 waves (wave32)
#define TILE_FLOATS    (TOK_PER_BLK * ROW)   // 12288 floats = 48 KB

__device__ __forceinline__ int pack16(int lo, int hi) {
    return (lo & 0xFFFF) | (hi << 16);
}

__global__ __launch_bounds__(BLK_THREADS)
void embedding_gather_tdm(const int* __restrict__ ids,
                          const float* __restrict__ weight,
                          float* __restrict__ out) {
    __shared__ __align__(16) vf4 tile[TILE_FLOATS / 4];   // only LDS object -> lds_addr 0

    const int tok0 = blockIdx.x * TOK_PER_BLK;

    if (threadIdx.x < 32) {   // wave 0 drives the TDM
        // 16 wave-uniform row indices, pinned to SGPRs for the descriptor.
        int r[TOK_PER_BLK];
#pragma unroll
        for (int j = 0; j < TOK_PER_BLK; ++j)
            r[j] = __builtin_amdgcn_readfirstlane(ids[tok0 + j]);

        // ---- D# group 0 (128b): count=1, gather_mode=1, 16-bit indices,
        //      lds_addr=0, global_addr=weight base, type=2 ("image").
        const u64 ga = (u64)weight;
        v4u g0;
        g0[0] = 0x80000001u;                                   // gather_mode<<31 | count=1
        g0[1] = 0u;                                            // lds_addr (bytes)
        g0[2] = (u32)(ga & 0xFFFFFFFFull);                     // global_addr[31:0]
        g0[3] = (u32)((ga >> 32) & 0x01FFFFFFull) | 0x80000000u; // addr[56:32] | type=2

        // ---- D# group 1 (256b): data_size=4B; tensor 768 x 32000 (row-major,
        //      stride 768); tile_dim0=768 (row width); tile_dim1=16 valid indices.
        v8i g1;
        g1[0] = 0x00020000;        // workgroup_mask=0, data_size=2 (4 bytes)
        g1[1] = (ROW << 16);       // atomic_barrier_addr=0 | tensor_dim0.lo=768
        g1[2] = (32000 << 16);     // tensor_dim0.hi=0 | tensor_dim1.lo=32000
        g1[3] = (ROW << 16);       // tensor_dim1.hi=0 | tile_dim0=768
        g1[4] = TOK_PER_BLK;       // tile_dim1 = #valid gather indices (16)
        g1[5] = ROW;               // tensor_dim0_stride = 768
        g1[6] = 0;                 // tensor_dim1_stride (ignored in gather mode)
        g1[7] = 0;

        // ---- D# groups 2/3: sixteen 16-bit row indices.
        v4i g2, g3;
        g2[0] = pack16(r[0],  r[1]);  g2[1] = pack16(r[2],  r[3]);
        g2[2] = pack16(r[4],  r[5]);  g2[3] = pack16(r[6],  r[7]);
        g3[0] = pack16(r[8],  r[9]);  g3[1] = pack16(r[10], r[11]);
        g3[2] = pack16(r[12], r[13]); g3[3] = pack16(r[14], r[15]);

#if defined(__clang_major__) && __clang_major__ >= 23
        v8i gx = {0, 0, 0, 0, 0, 0, 0, 0};
        __builtin_amdgcn_tensor_load_to_lds(g0, g1, g2, g3, gx, 0);
#else
        __builtin_amdgcn_tensor_load_to_lds(g0, g1, g2, g3, 0);
#endif
        __builtin_amdgcn_s_wait_tensorcnt(0);   // tile resident in LDS
    }
    __syncthreads();

    // The TDM builtin carries no LDS alias info. Escape the LDS address into
    // an asm with a memory clobber: the optimizer must then assume the DMA
    // wrote `tile`, so the ds loads below are live (no volatile needed, which
    // keeps them in addrspace(3) -> ds_load_b128, and lets them be clustered).
    asm volatile("" : : "v"(&tile[0]) : "memory");

    // Contiguous 48 KB LDS -> contiguous 48 KB of output, b128 granularity.
    vf4* dst = (vf4*)(out + (size_t)tok0 * ROW);
#pragma unroll
    for (int i = 0; i < TILE_FLOATS / 4 / BLK_THREADS; ++i) {   // 12 iters
        const int k = threadIdx.x + i * BLK_THREADS;
        dst[k] = tile[k];
    }
}

extern "C" void kernel_launch(void* const* d_in, const int* in_sizes, int n_in,
                              void* d_out, int out_size, void* d_ws, size_t ws_size,
                              hipStream_t stream) {
    const int*   ids    = (const int*)d_in[0];     // input_ids [4,2048] int32
    const float* weight = (const float*)d_in[1];   // weight [32000,768] f32
    float*       out    = (float*)d_out;           // [4,2048,768] f32

    const int n_tok  = in_sizes[0];                // 8192
    const int blocks = n_tok / TOK_PER_BLK;        // 512

    embedding_gather_tdm<<<dim3(blocks), dim3(BLK_THREADS), 0, stream>>>(ids, weight, out);
}